// KernelConv2D_377957122763
// MI455X (gfx1250) — compile-verified
//
#include <hip/hip_runtime.h>
#include <stdint.h>

// Per-pixel dynamic 5x5 depthwise conv + leaky ReLU(0.2), replicate padding.
// feat:   (2, 32, 256, 256) f32
// kernel: (2, 32*25, 256, 256) f32   (tap index t = i*5+j within each c)
// out:    (2, 32, 256, 256) f32
//
// Memory-bound (~453 MB @ 23.3 TB/s ~= 19.4 us). Strategy:
//  - stream the once-used 419 MB kernel tensor with NT loads (don't pollute L2,
//    keep the 16.8 MB feat resident in the 192 MB L2)
//  - stage the 25x-reused feat tile into LDS with CDNA5 async global->LDS
//    copies (ASYNCcnt / s_wait_asynccnt), materializing replicate padding

#define KS      5
#define PADR    2
#define HH      256
#define WW      256
#define CC      32
#define NIMG    2
#define ROWS    16            // output rows per block
#define TROWS   (ROWS + 4)    // staged feat rows (vertical halo, clamped)
#define TSTRIDE 264           // floats per staged LDS row
// LDS row layout: cols [4,260) = feat cols 0..255 (16B-aligned interior),
// cols 2,3 = left replicate pad, cols 260,261 = right replicate pad.
// Window col for output w, tap dj:  2 + w + dj   (range 2..261)

typedef float v4f __attribute__((ext_vector_type(4)));

__global__ __launch_bounds__(256) void dyn_dwconv5x5(
    const float* __restrict__ feat,
    const float* __restrict__ kern,
    float* __restrict__ out)
{
  __shared__ __align__(16) float tile[TROWS][TSTRIDE];

  const int tid   = threadIdx.x;
  const int strip = blockIdx.x & 15;       // 16 strips of 16 rows
  const int plane = blockIdx.x >> 4;       // n*C + c, in [0,64)
  const int r0    = strip * ROWS;

  const float* featp = feat + (size_t)plane * (HH * WW);
  const float* kernp = kern + (size_t)plane * 25 * (HH * WW);
  float*       outp  = out  + (size_t)plane * (HH * WW);

  // ---- Stage feat strip rows [r0-2, r0+17] (clamped) into LDS via CDNA5
  // async global->LDS b128 copies: 20 rows x 64 float4 = 1280 = 5 x 256 lanes.
  const uint32_t lds_base = (uint32_t)(uintptr_t)(&tile[0][0]);
#pragma unroll
  for (int it = 0; it < (TROWS * (WW / 4)) / 256; ++it) {
    const int idx  = it * 256 + tid;
    const int k    = idx >> 6;             // staged row 0..19
    const int q    = idx & 63;             // float4 column 0..63
    int grow = r0 - PADR + k;
    grow = grow < 0 ? 0 : (grow > HH - 1 ? HH - 1 : grow);
    const uint64_t g = (uint64_t)(uintptr_t)(featp + grow * WW + q * 4);
    const uint32_t l = lds_base + (uint32_t)((k * TSTRIDE + 4 + q * 4) * sizeof(float));
    asm volatile("global_load_async_to_lds_b128 %0, %1, off"
                 :: "v"(l), "v"(g)
                 : "memory");
  }
  asm volatile("s_wait_asynccnt 0x0" ::: "memory");
  __syncthreads();

  // Materialize horizontal replicate padding in LDS.
  if (tid < TROWS) {
    const float lft = tile[tid][4];
    const float rgt = tile[tid][259];
    tile[tid][2]   = lft;  tile[tid][3]   = lft;
    tile[tid][260] = rgt;  tile[tid][261] = rgt;
  }
  __syncthreads();

  // ---- Compute: each thread owns 4 consecutive pixels in each of 4 rows.
  const int w4 = (tid & 63) * 4;   // first output col
  const int ty = tid >> 6;         // 0..3

#pragma unroll
  for (int rit = 0; rit < 4; ++rit) {
    const int rr = ty + rit * 4;   // local output row 0..15
    const int h  = r0 + rr;
    v4f acc = {0.f, 0.f, 0.f, 0.f};
#pragma unroll
    for (int di = 0; di < KS; ++di) {
      // 8-float sliding window from LDS (covers dj=0..4 for 4 pixels)
      float f[8];
#pragma unroll
      for (int x = 0; x < 8; ++x) f[x] = tile[rr + di][2 + w4 + x];
#pragma unroll
      for (int dj = 0; dj < KS; ++dj) {
        const int t = di * KS + dj;
        const v4f kv = __builtin_nontemporal_load(
            (const v4f*)(kernp + (size_t)t * (HH * WW) + h * WW + w4));
        acc.x = __builtin_fmaf(f[dj + 0], kv.x, acc.x);
        acc.y = __builtin_fmaf(f[dj + 1], kv.y, acc.y);
        acc.z = __builtin_fmaf(f[dj + 2], kv.z, acc.z);
        acc.w = __builtin_fmaf(f[dj + 3], kv.w, acc.w);
      }
    }
    // leaky ReLU, slope 0.2
    acc.x = acc.x >= 0.f ? acc.x : 0.2f * acc.x;
    acc.y = acc.y >= 0.f ? acc.y : 0.2f * acc.y;
    acc.z = acc.z >= 0.f ? acc.z : 0.2f * acc.z;
    acc.w = acc.w >= 0.f ? acc.w : 0.2f * acc.w;
    __builtin_nontemporal_store(acc, (v4f*)(outp + h * WW + w4));
  }
}

extern "C" void kernel_launch(void* const* d_in, const int* in_sizes, int n_in,
                              void* d_out, int out_size, void* d_ws, size_t ws_size,
                              hipStream_t stream) {
  (void)in_sizes; (void)n_in; (void)out_size; (void)d_ws; (void)ws_size;
  const float* feat = (const float*)d_in[0];
  const float* kern = (const float*)d_in[1];
  float*       o    = (float*)d_out;
  const int nblocks = NIMG * CC * (HH / ROWS);   // 1024
  dyn_dwconv5x5<<<dim3(nblocks), dim3(256), 0, stream>>>(feat, kern, o);
}